// GATConcDQL_78417512890495
// MI455X (gfx1250) — compile-verified
//
#include <hip/hip_runtime.h>

typedef __attribute__((ext_vector_type(2))) float v2f;
typedef __attribute__((ext_vector_type(8))) float v8f;

#define TPB 256

__device__ __forceinline__ unsigned ford(float f){
  unsigned u = __float_as_uint(f);
  return (u & 0x80000000u) ? ~u : (u | 0x80000000u);
}
__device__ __forceinline__ float fdec(unsigned u){
  return (u & 0x80000000u) ? __uint_as_float(u & 0x7fffffffu) : __uint_as_float(~u);
}

// ---------------- zero small accumulators ----------------
__global__ void k_zero(float* p, int n){
  for(int i=blockIdx.x*blockDim.x+threadIdx.x; i<n; i+=gridDim.x*blockDim.x) p[i]=0.f;
}

// ---------------- GraphNorm ----------------
__global__ void k_gn_sum(const float* __restrict__ x, const int* __restrict__ batch,
                         float* gsum, float* cntf, int N){
  __shared__ float s[8];
  int tid=threadIdx.x;
  if(tid<8) s[tid]=0.f;
  __syncthreads();
  int start = blockIdx.x*blockDim.x;
  int i = start + tid;
  int g0 = batch[start];
  if(i<N){
    const float4* x4 = (const float4*)x;
    float4 a = x4[(size_t)i*2+0];
    float4 b = x4[(size_t)i*2+1];
    atomicAdd(&s[0], a.x); atomicAdd(&s[1], a.y); atomicAdd(&s[2], a.z); atomicAdd(&s[3], a.w);
    atomicAdd(&s[4], b.x); atomicAdd(&s[5], b.y); atomicAdd(&s[6], b.z); atomicAdd(&s[7], b.w);
  }
  __syncthreads();
  if(tid<8) atomicAdd(&gsum[g0*8+tid], s[tid]);
  if(tid==0){
    int act = N - start; if(act > (int)blockDim.x) act = (int)blockDim.x;
    atomicAdd(&cntf[g0], (float)act);
  }
}

__global__ void k_gn_var(const float* __restrict__ x, const int* __restrict__ batch,
                         const float* __restrict__ gsum, const float* __restrict__ cntf,
                         const float* __restrict__ ms, float* gvar, int N){
  __shared__ float s[8];
  int tid=threadIdx.x;
  if(tid<8) s[tid]=0.f;
  __syncthreads();
  int start = blockIdx.x*blockDim.x;
  int i = start + tid;
  int g0 = batch[start];
  float inv = 1.f/cntf[g0];
  if(i<N){
    #pragma unroll
    for(int f=0;f<8;++f){
      float sub = x[(size_t)i*8+f] - ms[f]*(gsum[g0*8+f]*inv);
      atomicAdd(&s[f], sub*sub);
    }
  }
  __syncthreads();
  if(tid<8) atomicAdd(&gvar[g0*8+tid], s[tid]);
}

__global__ void k_gn_out(const float* __restrict__ x, const int* __restrict__ batch,
                         const float* __restrict__ gsum, const float* __restrict__ gvar,
                         const float* __restrict__ cntf, const float* __restrict__ w,
                         const float* __restrict__ b, const float* __restrict__ ms,
                         float* __restrict__ xn, int N){
  int i = blockIdx.x*blockDim.x + threadIdx.x;
  if(i>=N) return;
  int g = batch[i];
  float inv = 1.f/cntf[g];
  float o[8];
  #pragma unroll
  for(int f=0;f<8;++f){
    float sub = x[(size_t)i*8+f] - ms[f]*(gsum[g*8+f]*inv);
    float var = gvar[g*8+f]*inv;
    o[f] = w[f]*sub*rsqrtf(var + 1e-5f) + b[f];
  }
  float4* xn4 = (float4*)xn;
  xn4[(size_t)i*2+0] = make_float4(o[0],o[1],o[2],o[3]);
  xn4[(size_t)i*2+1] = make_float4(o[4],o[5],o[6],o[7]);
}

// ---------------- edge_attr column mean ----------------
__global__ void k_ea_sum(const float* __restrict__ ea, float* eaSum, int E){
  const float4* ea4 = (const float4*)ea;
  float4 loc = make_float4(0.f,0.f,0.f,0.f);
  for(int e=blockIdx.x*blockDim.x+threadIdx.x; e<E; e+=gridDim.x*blockDim.x){
    float4 a = ea4[e];
    loc.x+=a.x; loc.y+=a.y; loc.z+=a.z; loc.w+=a.w;
  }
  __shared__ float s[4];
  if(threadIdx.x<4) s[threadIdx.x]=0.f;
  __syncthreads();
  atomicAdd(&s[0], loc.x); atomicAdd(&s[1], loc.y);
  atomicAdd(&s[2], loc.z); atomicAdd(&s[3], loc.w);
  __syncthreads();
  if(threadIdx.x<4) atomicAdd(&eaSum[threadIdx.x], s[threadIdx.x]);
}

// ---------------- WMMA node transforms: out(stride 8) = x @ W + b  (W is [KIN,5]) ----------------
// Fragment layouts per CDNA5 ISA 7.12.2:
//   A (16x4 f32, 2 VGPRs): lane m=lane&15, k = j + 2*(lane>>4) within chunk
//   B (4x16 f32, 2 VGPRs): lane n=lane&15, row k = j + 2*(lane>>4) within chunk
//   D (16x16 f32, 8 VGPRs): element (m = v + 8*(lane>>4), n = lane&15)
template<int KIN, int XS>
__global__ void k_transform(const float* __restrict__ x,
                            const float* __restrict__ Wl, const float* __restrict__ bl,
                            const float* __restrict__ Wr, const float* __restrict__ br,
                            float* __restrict__ xl, float* __restrict__ xr, int nTiles){
  int wavesPerBlock = blockDim.x >> 5;
  int tile = blockIdx.x*wavesPerBlock + (threadIdx.x >> 5);
  if(tile >= nTiles) return;
  int lane = threadIdx.x & 31;
  int hw = lane >> 4;
  int n  = lane & 15;
  int rowBase = tile*16;
  constexpr int CH = (KIN + 3)/4;
  v2f a[CH], bL[CH], bR[CH];
  int row = rowBase + n;     // A-matrix row for this lane
  #pragma unroll
  for(int c=0;c<CH;++c){
    int k0 = c*4 + 2*hw;
    a[c].x = (k0   < XS) ? x[(size_t)row*XS + k0]     : 0.f;
    a[c].y = (k0+1 < XS) ? x[(size_t)row*XS + k0 + 1] : 0.f;
    float w0l=0.f,w1l=0.f,w0r=0.f,w1r=0.f;
    if(n < 5){
      if(k0   < KIN){ w0l = Wl[k0*5+n];     w0r = Wr[k0*5+n]; }
      if(k0+1 < KIN){ w1l = Wl[(k0+1)*5+n]; w1r = Wr[(k0+1)*5+n]; }
    }
    bL[c].x=w0l; bL[c].y=w1l; bR[c].x=w0r; bR[c].y=w1r;
  }
  v8f cl = {}; v8f cr = {};
  #pragma unroll
  for(int c=0;c<CH;++c){
    cl = __builtin_amdgcn_wmma_f32_16x16x4_f32(false, a[c], false, bL[c], (short)0, cl, false, false);
    cr = __builtin_amdgcn_wmma_f32_16x16x4_f32(false, a[c], false, bR[c], (short)0, cr, false, false);
  }
  if(n < 8){
    float bln = (n<5) ? bl[n] : 0.f;
    float brn = (n<5) ? br[n] : 0.f;
    #pragma unroll
    for(int v=0; v<8; ++v){
      int m = v + 8*hw;
      float vl = (n<5) ? cl[v] + bln : 0.f;   // zero-fill pad lanes 5..7
      float vr = (n<5) ? cr[v] + brn : 0.f;
      xl[(size_t)(rowBase+m)*8 + n] = vl;
      xr[(size_t)(rowBase+m)*8 + n] = vr;
    }
  }
}

// ---------------- GAT per-layer node prep: self-loop logit, init accumulators ----------------
__global__ void k_prep(const float* __restrict__ xl, const float* __restrict__ xr,
                       const float* __restrict__ We, const float* __restrict__ att,
                       const float* __restrict__ eaSum, float invE,
                       float* __restrict__ lself, unsigned* __restrict__ mOrd,
                       float* __restrict__ denom, float* __restrict__ num, int N){
  int i = blockIdx.x*blockDim.x + threadIdx.x;
  if(i>=N) return;
  const float4* xl4 = (const float4*)xl;
  const float4* xr4 = (const float4*)xr;
  float4 la = xl4[(size_t)i*2];  float l4 = xl[(size_t)i*8+4];
  float4 ra = xr4[(size_t)i*2];  float r4 = xr[(size_t)i*8+4];
  float m0=eaSum[0]*invE, m1=eaSum[1]*invE, m2=eaSum[2]*invE, m3=eaSum[3]*invE;
  float sl[5] = { la.x+ra.x, la.y+ra.y, la.z+ra.z, la.w+ra.w, l4+r4 };
  float logit = 0.f;
  #pragma unroll
  for(int j=0;j<5;++j){
    float v = sl[j] + m0*We[j] + m1*We[5+j] + m2*We[10+j] + m3*We[15+j];
    v = v>0.f ? v : 0.2f*v;
    logit += att[j]*v;
  }
  lself[i] = logit;
  mOrd[i]  = ford(logit);
  denom[i] = 0.f;
  float4* num4 = (float4*)num;
  num4[(size_t)i*2+0] = make_float4(0.f,0.f,0.f,0.f);
  num4[(size_t)i*2+1] = make_float4(0.f,0.f,0.f,0.f);
}

// ---------------- edge pass 1: logits + segment max ----------------
__global__ void k_edge1(const int* __restrict__ ei, const float* __restrict__ ea,
                        const float* __restrict__ xl, const float* __restrict__ xr,
                        const float* __restrict__ We, const float* __restrict__ att,
                        float* __restrict__ logits, unsigned* __restrict__ mOrd, int E){
  int e = blockIdx.x*blockDim.x + threadIdx.x;
  if(e>=E) return;
  int s = ei[e], d = ei[(size_t)E + e];
  float4 av = ((const float4*)ea)[e];
  const float4* xl4 = (const float4*)xl;
  const float4* xr4 = (const float4*)xr;
  float4 ls = xl4[(size_t)s*2];  float ls4 = xl[(size_t)s*8+4];
  float4 rd = xr4[(size_t)d*2];  float rd4 = xr[(size_t)d*8+4];
  float sl[5] = { ls.x+rd.x, ls.y+rd.y, ls.z+rd.z, ls.w+rd.w, ls4+rd4 };
  float logit = 0.f;
  #pragma unroll
  for(int j=0;j<5;++j){
    float v = sl[j] + av.x*We[j] + av.y*We[5+j] + av.z*We[10+j] + av.w*We[15+j];
    v = v>0.f ? v : 0.2f*v;
    logit += att[j]*v;
  }
  logits[e] = logit;
  atomicMax(&mOrd[d], ford(logit));
}

// ---------------- edge pass 2: softmax numerator/denominator ----------------
__global__ void k_edge2(const int* __restrict__ ei, const float* __restrict__ logits,
                        const float* __restrict__ xl, const unsigned* __restrict__ mOrd,
                        float* __restrict__ denom, float* __restrict__ num, int E){
  int e = blockIdx.x*blockDim.x + threadIdx.x;
  if(e>=E) return;
  int s = ei[e], d = ei[(size_t)E + e];
  float ex = __expf(logits[e] - fdec(mOrd[d]));
  const float4* xl4 = (const float4*)xl;
  float4 ls = xl4[(size_t)s*2];
  float ls4 = xl[(size_t)s*8+4];
  atomicAdd(&denom[d], ex);
  atomicAdd(&num[(size_t)d*8+0], ex*ls.x);
  atomicAdd(&num[(size_t)d*8+1], ex*ls.y);
  atomicAdd(&num[(size_t)d*8+2], ex*ls.z);
  atomicAdd(&num[(size_t)d*8+3], ex*ls.w);
  atomicAdd(&num[(size_t)d*8+4], ex*ls4);
}

// ---------------- finalize layer 1: h=relu(agg)+concat -> x13 (stride 16, zero padded) ----------------
__global__ void k_fin1(const unsigned* __restrict__ mOrd, const float* __restrict__ lself,
                       const float* __restrict__ denom, const float* __restrict__ num,
                       const float* __restrict__ xl, const float* __restrict__ bo,
                       const float* __restrict__ xn, float* __restrict__ x13, int N){
  int i = blockIdx.x*blockDim.x + threadIdx.x;
  if(i>=N) return;
  float exs = __expf(lself[i] - fdec(mOrd[i]));
  float invd = 1.f/(denom[i] + exs);
  const float4* num4 = (const float4*)num;
  const float4* xl4  = (const float4*)xl;
  float4 n0 = num4[(size_t)i*2];  float n4 = num[(size_t)i*8+4];
  float4 l0 = xl4[(size_t)i*2];   float l4 = xl[(size_t)i*8+4];
  float h[5];
  h[0]=(n0.x+exs*l0.x)*invd+bo[0]; h[1]=(n0.y+exs*l0.y)*invd+bo[1];
  h[2]=(n0.z+exs*l0.z)*invd+bo[2]; h[3]=(n0.w+exs*l0.w)*invd+bo[3];
  h[4]=(n4  +exs*l4  )*invd+bo[4];
  const float4* xn4 = (const float4*)xn;
  float4 xa = xn4[(size_t)i*2+0];
  float4 xb = xn4[(size_t)i*2+1];
  float4* o4 = (float4*)x13;
  o4[(size_t)i*4+0] = make_float4(fmaxf(h[0],0.f), fmaxf(h[1],0.f), fmaxf(h[2],0.f), fmaxf(h[3],0.f));
  o4[(size_t)i*4+1] = make_float4(fmaxf(h[4],0.f), xa.x, xa.y, xa.z);
  o4[(size_t)i*4+2] = make_float4(xa.w, xb.x, xb.y, xb.z);
  o4[(size_t)i*4+3] = make_float4(xb.w, 0.f, 0.f, 0.f);
}

// ---------------- finalize layer 2 -> x18 (stride 18) ----------------
__global__ void k_fin2(const unsigned* __restrict__ mOrd, const float* __restrict__ lself,
                       const float* __restrict__ denom, const float* __restrict__ num,
                       const float* __restrict__ xl, const float* __restrict__ bo,
                       const float* __restrict__ x13, float* __restrict__ x18, int N){
  int i = blockIdx.x*blockDim.x + threadIdx.x;
  if(i>=N) return;
  float exs = __expf(lself[i] - fdec(mOrd[i]));
  float invd = 1.f/(denom[i] + exs);
  const float4* num4 = (const float4*)num;
  const float4* xl4  = (const float4*)xl;
  float4 n0 = num4[(size_t)i*2];  float n4 = num[(size_t)i*8+4];
  float4 l0 = xl4[(size_t)i*2];   float l4 = xl[(size_t)i*8+4];
  x18[(size_t)i*18+0] = fmaxf((n0.x+exs*l0.x)*invd+bo[0], 0.f);
  x18[(size_t)i*18+1] = fmaxf((n0.y+exs*l0.y)*invd+bo[1], 0.f);
  x18[(size_t)i*18+2] = fmaxf((n0.z+exs*l0.z)*invd+bo[2], 0.f);
  x18[(size_t)i*18+3] = fmaxf((n0.w+exs*l0.w)*invd+bo[3], 0.f);
  x18[(size_t)i*18+4] = fmaxf((n4  +exs*l4  )*invd+bo[4], 0.f);
  #pragma unroll
  for(int f=0;f<13;++f) x18[(size_t)i*18+5+f] = x13[(size_t)i*16+f];
}

// ---------------- per-graph mean pool (block-local LDS reduce, one block = one graph slice) ----------------
__global__ void k_pool(const float* __restrict__ x18, const int* __restrict__ batch,
                       float* pool, int N){
  __shared__ float s[18];
  int tid = threadIdx.x;
  if(tid<18) s[tid]=0.f;
  __syncthreads();
  int start = blockIdx.x*blockDim.x;
  int i = start + tid;
  int g0 = batch[start];
  if(i<N){
    #pragma unroll
    for(int f=0;f<18;++f) atomicAdd(&s[f], x18[(size_t)i*18+f]);
  }
  __syncthreads();
  if(tid<18) atomicAdd(&pool[g0*18+tid], s[tid]);
}

// ---------------- dueling head ----------------
__global__ void k_head(const float* __restrict__ x18, const float* __restrict__ pool,
                       const float* __restrict__ cntf, const int* __restrict__ cur,
                       const int* __restrict__ amask, const float* __restrict__ goal,
                       const float* __restrict__ Wv1, const float* __restrict__ bv1,
                       const float* __restrict__ Wv2, const float* __restrict__ bv2,
                       const float* __restrict__ Wa1, const float* __restrict__ ba1,
                       const float* __restrict__ Wa2, const float* __restrict__ ba2,
                       float* __restrict__ out, int B){
  int g = blockIdx.x*blockDim.x + threadIdx.x;
  if(g>=B) return;
  int off = 0;
  for(int k=0;k<g;++k) off += (int)(cntf[k] + 0.5f);
  int node = cur[g] + off;
  float feat[42];
  #pragma unroll
  for(int f=0;f<18;++f) feat[f] = x18[(size_t)node*18+f];
  float invc = 1.f/cntf[g];
  #pragma unroll
  for(int f=0;f<18;++f) feat[18+f] = pool[g*18+f]*invc;
  #pragma unroll
  for(int f=0;f<6;++f)  feat[36+f] = goal[g*6+f];

  float hid[10];
  for(int h=0;h<10;++h){
    float a = ba1[h];
    for(int i=0;i<42;++i) a += feat[i]*Wa1[i*10+h];
    hid[h] = fmaxf(a, 0.f);
  }
  float adv[4];
  for(int j=0;j<4;++j){
    float a = ba2[j];
    for(int h=0;h<10;++h) a += hid[h]*Wa2[h*4+j];
    adv[j] = a;
  }
  float val = bv2[0];
  for(int h=0;h<10;++h){
    float vh = bv1[h];
    for(int i=0;i<42;++i) vh += feat[i]*Wv1[i*10+h];
    val += fmaxf(vh, 0.f)*Wv2[h];
  }
  float am = (adv[0]+adv[1]+adv[2]+adv[3])*0.25f;
  for(int j=0;j<4;++j){
    float q = val + adv[j] - am;
    out[g*4+j] = (amask[g*4+j]==0) ? -1e8f : q;
  }
}

extern "C" void kernel_launch(void* const* d_in, const int* in_sizes, int n_in,
                              void* d_out, int out_size, void* d_ws, size_t ws_size,
                              hipStream_t stream){
  const float* x     = (const float*)d_in[0];
  const int*   ei    = (const int*)  d_in[1];
  const float* ea    = (const float*)d_in[2];
  const int*   batch = (const int*)  d_in[3];
  const int*   cur   = (const int*)  d_in[4];
  const int*   amask = (const int*)  d_in[5];
  const float* goal  = (const float*)d_in[6];
  const float* gn_w  = (const float*)d_in[7];
  const float* gn_b  = (const float*)d_in[8];
  const float* gn_ms = (const float*)d_in[9];
  const float* Wl1=(const float*)d_in[10], *bl1=(const float*)d_in[11];
  const float* Wr1=(const float*)d_in[12], *br1=(const float*)d_in[13];
  const float* We1=(const float*)d_in[14], *att1=(const float*)d_in[15], *bo1=(const float*)d_in[16];
  const float* Wl2=(const float*)d_in[17], *bl2=(const float*)d_in[18];
  const float* Wr2=(const float*)d_in[19], *br2=(const float*)d_in[20];
  const float* We2=(const float*)d_in[21], *att2=(const float*)d_in[22], *bo2=(const float*)d_in[23];
  const float* Wv1=(const float*)d_in[24], *bv1=(const float*)d_in[25];
  const float* Wv2=(const float*)d_in[26], *bv2=(const float*)d_in[27];
  const float* Wa1=(const float*)d_in[28], *ba1=(const float*)d_in[29];
  const float* Wa2=(const float*)d_in[30], *ba2=(const float*)d_in[31];

  const int N = in_sizes[0]/8;
  const int E = in_sizes[1]/2;
  const int B = in_sizes[4];

  // ---- workspace carve-up (all f32; xl/xr/num padded to stride 8) ----
  float* W = (float*)d_ws;
  size_t o = 0;
  float* xn    = W+o; o += (size_t)N*8;
  float* x13   = W+o; o += (size_t)N*16;
  float* x18   = W+o; o += (size_t)N*18;
  float* xl    = W+o; o += (size_t)N*8;
  float* xr    = W+o; o += (size_t)N*8;
  float* lself = W+o; o += (size_t)N;
  unsigned* mOrd = (unsigned*)(W+o); o += (size_t)N;
  float* denom = W+o; o += (size_t)N;
  float* num   = W+o; o += (size_t)N*8;
  float* logits= W+o; o += (size_t)E;
  float* gsum  = W+o; o += 512;
  float* gvar  = W+o; o += 512;
  float* cntf  = W+o; o += 64;
  float* eaSum = W+o; o += 4;
  float* pool  = W+o; o += 64*18;
  const int smallN = 512+512+64+4+64*18;

  const int nodeBlocks = (N + TPB - 1)/TPB;
  const int edgeBlocks = (E + TPB - 1)/TPB;
  const int nTiles = (N + 15)/16;
  const int tileBlocks = (nTiles + (TPB/32) - 1)/(TPB/32);
  const float invE = 1.f/(float)E;

  // 0) zero small accumulators (gsum..pool contiguous)
  k_zero<<<8, TPB, 0, stream>>>(gsum, smallN);

  // 1) GraphNorm
  k_gn_sum<<<nodeBlocks, TPB, 0, stream>>>(x, batch, gsum, cntf, N);
  k_gn_var<<<nodeBlocks, TPB, 0, stream>>>(x, batch, gsum, cntf, gn_ms, gvar, N);
  k_gn_out<<<nodeBlocks, TPB, 0, stream>>>(x, batch, gsum, gvar, cntf, gn_w, gn_b, gn_ms, xn, N);

  // 2) edge_attr mean
  k_ea_sum<<<2048, TPB, 0, stream>>>(ea, eaSum, E);

  // 3) GAT layer 1
  k_transform<8,8><<<tileBlocks, TPB, 0, stream>>>(xn, Wl1, bl1, Wr1, br1, xl, xr, nTiles);
  k_prep<<<nodeBlocks, TPB, 0, stream>>>(xl, xr, We1, att1, eaSum, invE, lself, mOrd, denom, num, N);
  k_edge1<<<edgeBlocks, TPB, 0, stream>>>(ei, ea, xl, xr, We1, att1, logits, mOrd, E);
  k_edge2<<<edgeBlocks, TPB, 0, stream>>>(ei, logits, xl, mOrd, denom, num, E);
  k_fin1<<<nodeBlocks, TPB, 0, stream>>>(mOrd, lself, denom, num, xl, bo1, xn, x13, N);

  // 4) GAT layer 2
  k_transform<13,16><<<tileBlocks, TPB, 0, stream>>>(x13, Wl2, bl2, Wr2, br2, xl, xr, nTiles);
  k_prep<<<nodeBlocks, TPB, 0, stream>>>(xl, xr, We2, att2, eaSum, invE, lself, mOrd, denom, num, N);
  k_edge1<<<edgeBlocks, TPB, 0, stream>>>(ei, ea, xl, xr, We2, att2, logits, mOrd, E);
  k_edge2<<<edgeBlocks, TPB, 0, stream>>>(ei, logits, xl, mOrd, denom, num, E);
  k_fin2<<<nodeBlocks, TPB, 0, stream>>>(mOrd, lself, denom, num, xl, bo2, x13, x18, N);

  // 5) pool + head
  k_pool<<<nodeBlocks, TPB, 0, stream>>>(x18, batch, pool, N);
  k_head<<<1, 64, 0, stream>>>(x18, pool, cntf, cur, amask, goal,
                               Wv1, bv1, Wv2, bv2, Wa1, ba1, Wa2, ba2,
                               (float*)d_out, B);
}